// GCN_51728586113457
// MI455X (gfx1250) — compile-verified
//
#include <hip/hip_runtime.h>
#include <hip/hip_bf16.h>

typedef float v2f __attribute__((ext_vector_type(2)));
typedef float v8f __attribute__((ext_vector_type(8)));

#define N_NODES  50000
#define N_EDGES  600000
#define N_GRAPHS 64
#define D_IN     128
#define D_HID    128
#define D_OUT    64

// ---------------------------------------------------------------------------
// H[N,F] = X[N,K] * W[K,F]  via V_WMMA_F32_16X16X4_F32 (exact fp32).
// One wave (32 threads) per 16x16 output tile.
// A frag (16x4 f32, 2 VGPR/lane): lane l -> row m = l&15, k-pair = 2*(l>>4).
// B frag (4x16 f32, 2 VGPR/lane): lane l -> col n = l&15, k-pair = 2*(l>>4).
// C/D  (16x16 f32, 8 VGPR/lane):  VGPR i -> M = 8*(l>>4)+i, N = l&15.
// ---------------------------------------------------------------------------
__global__ __launch_bounds__(32)
void gemm_wmma_f32(const float* __restrict__ X, const float* __restrict__ W,
                   float* __restrict__ H, int K, int F) {
  const int rowTile = blockIdx.x << 4;
  const int colTile = blockIdx.y << 4;
  const int lane = threadIdx.x & 31;
  const int half = lane >> 4;      // 0 or 1
  const int i16  = lane & 15;

  const float* xrow = X + (size_t)(rowTile + i16) * (size_t)K;
  const float* wcol = W + colTile + i16;

  v8f acc = {};
  for (int k0 = 0; k0 < K; k0 += 4) {
    const int ka = k0 + (half << 1);
    v2f a = *(const v2f*)(xrow + ka);          // X[m][ka], X[m][ka+1]
    v2f b;
    b.x = wcol[(size_t)ka * (size_t)F];        // W[ka  ][n]
    b.y = wcol[(size_t)(ka + 1) * (size_t)F];  // W[ka+1][n]
    acc = __builtin_amdgcn_wmma_f32_16x16x4_f32(
        /*neg_a=*/false, a, /*neg_b=*/false, b,
        /*c_mod=*/(short)0, acc, /*reuse_a=*/false, /*reuse_b=*/false);
  }

  float* out = H + (size_t)(rowTile + (half << 3)) * (size_t)F + colTile + i16;
#pragma unroll
  for (int i = 0; i < 8; ++i) out[(size_t)i * (size_t)F] = acc[i];
}

// ---------------------------------------------------------------------------
__global__ void zero_f32(float* __restrict__ p, int n) {
  int i = blockIdx.x * blockDim.x + threadIdx.x;
  if (i < n) p[i] = 0.0f;
}

__global__ void deg_accum(const int* __restrict__ dst, float* __restrict__ deg, int E) {
  int e = blockIdx.x * blockDim.x + threadIdx.x;
  if (e < E) atomicAdd(&deg[dst[e]], 1.0f);
}

__global__ void deg_finalize(float* __restrict__ deg, int n) {
  int i = blockIdx.x * blockDim.x + threadIdx.x;
  if (i < n) deg[i] = rsqrtf(deg[i] + 1.0f);
}

// agg[i,f] = h[i,f] * dis[i]^2 + bias[f]   (self-loop + bias seed)
__global__ void agg_init(const float* __restrict__ h, const float* __restrict__ dis,
                         const float* __restrict__ bias, float* __restrict__ agg,
                         int F, int total) {
  int i = blockIdx.x * blockDim.x + threadIdx.x;
  if (i >= total) return;
  int node = i / F;
  int f = i - node * F;
  float d = dis[node];
  agg[i] = h[i] * d * d + bias[f];
}

// agg[dst,f] += h[src,f] * dis[src]*dis[dst]
__global__ void edge_scatter(const int* __restrict__ src, const int* __restrict__ dst,
                             const float* __restrict__ h, const float* __restrict__ dis,
                             float* __restrict__ agg, int F, int total) {
  int i = blockIdx.x * blockDim.x + threadIdx.x;
  if (i >= total) return;
  int e = i / F;
  int f = i - e * F;
  int s = src[e];
  int d = dst[e];
  float nm = dis[s] * dis[d];
  atomicAdd(&agg[(size_t)d * (size_t)F + f], h[(size_t)s * (size_t)F + f] * nm);
}

__global__ void relu_k(float* __restrict__ p, int n) {
  int i = blockIdx.x * blockDim.x + threadIdx.x;
  if (i < n) p[i] = fmaxf(p[i], 0.0f);
}

// sums[g,f] += h[i,f]; cnts[g] += 1 (once per node)
__global__ void pool_accum(const float* __restrict__ h, const int* __restrict__ batch,
                           float* __restrict__ sums, float* __restrict__ cnts, int total) {
  int i = blockIdx.x * blockDim.x + threadIdx.x;
  if (i >= total) return;
  int node = i >> 6;           // /64
  int f = i & 63;
  int g = batch[node];
  atomicAdd(&sums[(g << 6) + f], h[i]);
  if (f == 0) atomicAdd(&cnts[g], 1.0f);
}

__global__ void pool_final(const float* __restrict__ sums, const float* __restrict__ cnts,
                           float* __restrict__ out) {
  int i = blockIdx.x * blockDim.x + threadIdx.x;
  if (i < N_GRAPHS * D_OUT) {
    int g = i >> 6;
    out[i] = sums[i] / fmaxf(cnts[g], 1.0f);
  }
}

// ---------------------------------------------------------------------------
extern "C" void kernel_launch(void* const* d_in, const int* in_sizes, int n_in,
                              void* d_out, int out_size, void* d_ws, size_t ws_size,
                              hipStream_t stream) {
  const float* x     = (const float*)d_in[0];
  const int*   ei    = (const int*)d_in[1];   // [2, E]: src then dst
  const int*   batch = (const int*)d_in[2];
  const float* W1 = (const float*)d_in[3];  const float* b1 = (const float*)d_in[4];
  const float* W2 = (const float*)d_in[5];  const float* b2 = (const float*)d_in[6];
  const float* W3 = (const float*)d_in[7];  const float* b3 = (const float*)d_in[8];
  const float* W4 = (const float*)d_in[9];  const float* b4 = (const float*)d_in[10];
  float* out = (float*)d_out;

  float* bufA = (float*)d_ws;                                   // [N,128] gemm out
  float* bufB = bufA + (size_t)N_NODES * D_HID;                 // [N,128] agg / h
  float* dis  = bufB + (size_t)N_NODES * D_HID;                 // [N] deg_inv_sqrt
  float* sums = dis + N_NODES;                                  // [64,64]
  float* cnts = sums + N_GRAPHS * D_OUT;                        // [64]

  const int* src = ei;
  const int* dst = ei + N_EDGES;

  const int TB = 256;
  auto blocks = [](long long n, int tb) { return (unsigned)((n + tb - 1) / tb); };

  // degree -> deg_inv_sqrt (in place in dis)
  zero_f32<<<blocks(N_NODES, TB), TB, 0, stream>>>(dis, N_NODES);
  deg_accum<<<blocks(N_EDGES, TB), TB, 0, stream>>>(dst, dis, N_EDGES);
  deg_finalize<<<blocks(N_NODES, TB), TB, 0, stream>>>(dis, N_NODES);

  // one GCN layer: hin --(W,b)--> bufB (agg result), tmp = bufA
  auto layer = [&](const float* hin, int K, const float* W, const float* b, int F,
                   bool do_relu) {
    dim3 grid(N_NODES / 16, F / 16);
    gemm_wmma_f32<<<grid, 32, 0, stream>>>(hin, W, bufA, K, F);
    int totalNF = N_NODES * F;
    agg_init<<<blocks(totalNF, TB), TB, 0, stream>>>(bufA, dis, b, bufB, F, totalNF);
    int totalEF = N_EDGES * F;  // max 76.8M, fits int32
    edge_scatter<<<blocks(totalEF, TB), TB, 0, stream>>>(src, dst, bufA, dis, bufB, F,
                                                         totalEF);
    if (do_relu) relu_k<<<blocks(totalNF, TB), TB, 0, stream>>>(bufB, totalNF);
  };

  layer(x,    D_IN,  W1, b1, D_HID, true);
  layer(bufB, D_HID, W2, b2, D_HID, true);
  layer(bufB, D_HID, W3, b3, D_HID, true);
  layer(bufB, D_HID, W4, b4, D_OUT, false);

  // global mean pool (sums and cnts are contiguous -> single zero pass)
  zero_f32<<<blocks(N_GRAPHS * D_OUT + N_GRAPHS, TB), TB, 0, stream>>>(
      sums, N_GRAPHS * D_OUT + N_GRAPHS);
  pool_accum<<<blocks((long long)N_NODES * D_OUT, TB), TB, 0, stream>>>(
      bufB, batch, sums, cnts, N_NODES * D_OUT);
  pool_final<<<blocks(N_GRAPHS * D_OUT, TB), TB, 0, stream>>>(sums, cnts, out);
}